// MPNN_66623532696007
// MI455X (gfx1250) — compile-verified
//
#include <hip/hip_runtime.h>

typedef __attribute__((ext_vector_type(2))) float v2f;
typedef __attribute__((ext_vector_type(4))) float v4f;
typedef __attribute__((ext_vector_type(8))) float v8f;
typedef __attribute__((ext_vector_type(4))) int   v4i;

#define H       128
#define LAYERS  3
#define NNODES  2048
#define NEDGES  8192
#define NDIM    32
#define EDIM    16
#define NGRAPH  128
#define BN_EPS  1e-5f

#define KCHUNK   16
#define NCHUNK_W (EDIM * H / KCHUNK)        // 128 chunks of the main K=2048
#define NCHUNK_B (H / KCHUNK)               // 8 chunks of the edge-bias K=128
#define NCHUNK   (NCHUNK_W + NCHUNK_B)      // 136

// ---- async global->LDS path (guarded; falls back to sync copy) -------------
#if defined(__has_builtin)
#if __has_builtin(__builtin_amdgcn_global_load_async_to_lds_b128) && \
    __has_builtin(__builtin_amdgcn_s_wait_asynccnt)
#define USE_ASYNC_LDS 1
#endif
#endif

#if defined(USE_ASYNC_LDS)
typedef __attribute__((address_space(1))) v4i* as1_v4i;
typedef __attribute__((address_space(3))) v4i* as3_v4i;
#endif

__device__ __forceinline__ void stage_B(const float* __restrict__ gsrc,
                                        float* dst, int tid) {
  // copy KCHUNK x H floats (8 KB) global -> LDS, 128-bit granules
#if defined(USE_ASYNC_LDS)
  as1_v4i g = (as1_v4i)gsrc;
  as3_v4i l = (as3_v4i)dst;
  for (int i = tid; i < KCHUNK * H / 4; i += 256)
    __builtin_amdgcn_global_load_async_to_lds_b128(g + i, l + i, 0, 0);
#else
  for (int i = tid; i < KCHUNK * H / 4; i += 256)
    ((v4f*)dst)[i] = ((const v4f*)gsrc)[i];
#endif
}

__device__ __forceinline__ void wait_stage() {
#if defined(USE_ASYNC_LDS)
  __builtin_amdgcn_s_wait_asynccnt(0);
#endif
}

__device__ __forceinline__ v8f wmma_f32(v2f a, v2f b, v8f c) {
  // V_WMMA_F32_16X16X4_F32 : D = A(16x4) * B(4x16) + C(16x16), all fp32
  return __builtin_amdgcn_wmma_f32_16x16x4_f32(
      /*neg_a=*/false, a, /*neg_b=*/false, b,
      /*c_mod=*/(short)0, c, /*reuse_a=*/false, /*reuse_b=*/false);
}

// ---------------------------------------------------------------- node embed
// h = x @ node_W + node_b   [2048,32]@[32,128]
__global__ void k_node_embed(const float* __restrict__ x, const float* __restrict__ W,
                             const float* __restrict__ b, float* __restrict__ h) {
  int lane = threadIdx.x & 31, wave = threadIdx.x >> 5;
  int half = lane >> 4, mr = lane & 15;
  int m0 = blockIdx.x * 16, n0 = wave * 16;
  v8f acc = {};
  for (int k = 0; k < NDIM; k += 4) {
    int ka = k + 2 * half;
    v2f a, bf;
    a.x = x[(m0 + mr) * NDIM + ka];
    a.y = x[(m0 + mr) * NDIM + ka + 1];
    bf.x = W[ka * H + n0 + mr];
    bf.y = W[(ka + 1) * H + n0 + mr];
    acc = wmma_f32(a, bf, acc);
  }
  #pragma unroll
  for (int r = 0; r < 8; ++r) {
    int m = m0 + r + 8 * half, n = n0 + mr;
    h[m * H + n] = acc[r] + b[n];
  }
}

// ---------------------------------------------------------------- messages
// msg[e,:] = (kron(ea[e], h[src[e]])) @ edge_W_l  +  h[src[e]] @ edge_b_l
// Implicit A built in registers from LDS; B double-buffered via async->LDS.
__global__ void k_msg(const float* __restrict__ h, const int* __restrict__ edge_index,
                      const float* __restrict__ ea, const float* __restrict__ eW,
                      const float* __restrict__ eb, float* __restrict__ msg) {
  __shared__ float sh_h[64][H + 4];       // padded: conflict-free column reads
  __shared__ float sh_ea[64][EDIM + 4];
  __shared__ float sh_B[2][KCHUNK][H];    // ping-pong B panel
  __shared__ int   sh_src[64];

  int tid = threadIdx.x;
  int lane = tid & 31, wave = tid >> 5;
  int half = lane >> 4, mr = lane & 15;
  int wrow = wave & 3;       // which 16-edge tile
  int wcol = wave >> 2;      // which 64-col group
  int e0 = blockIdx.x * 64;

  // kick off first B chunk immediately (overlaps with A staging below)
  stage_B(eW, &sh_B[0][0][0], tid);

  if (tid < 64) sh_src[tid] = edge_index[e0 + tid];  // row 0 of edge_index = src
  __syncthreads();
  for (int idx = tid; idx < 64 * (H / 4); idx += 256) {
    int row = idx >> 5, c4 = idx & 31;
    *(v4f*)&sh_h[row][c4 * 4] = *(const v4f*)(h + (size_t)sh_src[row] * H + c4 * 4);
  }
  for (int idx = tid; idx < 64 * (EDIM / 4); idx += 256) {
    int row = idx >> 2, c4 = idx & 3;
    *(v4f*)&sh_ea[row][c4 * 4] = *(const v4f*)(ea + (size_t)(e0 + row) * EDIM + c4 * 4);
  }

  v8f acc[4] = {{}, {}, {}, {}};
  int m = wrow * 16 + mr;

  for (int c = 0; c < NCHUNK; ++c) {
    wait_stage();          // this wave's async writes for chunk c are in LDS
    __syncthreads();       // every wave's writes done; prior buffer fully read
    if (c + 1 < NCHUNK) {
      const float* nsrc = (c + 1 < NCHUNK_W)
          ? eW + (size_t)(c + 1) * KCHUNK * H
          : eb + (size_t)(c + 1 - NCHUNK_W) * KCHUNK * H;
      stage_B(nsrc, &sh_B[(c + 1) & 1][0][0], tid);
    }
    const float (*B)[H] = sh_B[c & 1];

    if (c < NCHUNK_W) {
      int kc = c * KCHUNK;
      #pragma unroll
      for (int kk = 0; kk < KCHUNK; kk += 4) {
        int kg = kc + kk;
        int d = kg >> 7;                 // edge_attr channel (const across 4-group)
        int kb = kk + 2 * half;          // k-row within chunk for this half-wave
        int ik = (kg + 2 * half) & 127;  // h column
        float ef = sh_ea[m][d];
        v2f a;
        a.x = ef * sh_h[m][ik];
        a.y = ef * sh_h[m][ik + 1];
        #pragma unroll
        for (int nt = 0; nt < 4; ++nt) {
          int n = wcol * 64 + nt * 16 + mr;
          v2f bf;
          bf.x = B[kb][n];
          bf.y = B[kb + 1][n];
          acc[nt] = wmma_f32(a, bf, acc[nt]);
        }
      }
    } else {
      int kloc = (c - NCHUNK_W) * KCHUNK;
      #pragma unroll
      for (int kk = 0; kk < KCHUNK; kk += 4) {
        int kb = kk + 2 * half;
        v2f a;
        a.x = sh_h[m][kloc + kb];
        a.y = sh_h[m][kloc + kb + 1];
        #pragma unroll
        for (int nt = 0; nt < 4; ++nt) {
          int n = wcol * 64 + nt * 16 + mr;
          v2f bf;
          bf.x = B[kb][n];
          bf.y = B[kb + 1][n];
          acc[nt] = wmma_f32(a, bf, acc[nt]);
        }
      }
    }
  }

  #pragma unroll
  for (int nt = 0; nt < 4; ++nt) {
    int n = wcol * 64 + nt * 16 + mr;
    #pragma unroll
    for (int r = 0; r < 8; ++r) {
      int e = e0 + wrow * 16 + r + 8 * half;
      msg[(size_t)e * H + n] = acc[nt][r];
    }
  }
}

// ---------------------------------------------------------------- scatter
__global__ void k_scatter_msg(const int* __restrict__ dst_idx, const float* __restrict__ msg,
                              float* __restrict__ agg, float* __restrict__ cnt) {
  int gid = blockIdx.x * blockDim.x + threadIdx.x;  // E*H
  int e = gid >> 7, c = gid & 127;
  int d = dst_idx[e];
  atomicAdd(&agg[(size_t)d * H + c], msg[gid]);
  if (c == 0) atomicAdd(&cnt[d], 1.0f);
}

// ---------------------------------------------------------------- root update
// x_new = h @ root_W + agg/max(cnt,1) + conv_bias
__global__ void k_update(const float* __restrict__ h, const float* __restrict__ rW,
                         const float* __restrict__ agg, const float* __restrict__ cnt,
                         const float* __restrict__ bias, float* __restrict__ x_new) {
  int lane = threadIdx.x & 31, wave = threadIdx.x >> 5;
  int half = lane >> 4, mr = lane & 15;
  int m0 = blockIdx.x * 16, n0 = wave * 16;
  v8f acc = {};
  for (int k = 0; k < H; k += 4) {
    int ka = k + 2 * half;
    v2f a, bf;
    a.x = h[(m0 + mr) * H + ka];
    a.y = h[(m0 + mr) * H + ka + 1];
    bf.x = rW[ka * H + n0 + mr];
    bf.y = rW[(ka + 1) * H + n0 + mr];
    acc = wmma_f32(a, bf, acc);
  }
  #pragma unroll
  for (int r = 0; r < 8; ++r) {
    int m = m0 + r + 8 * half, n = n0 + mr;
    float rc = 1.0f / fmaxf(cnt[m], 1.0f);
    x_new[(size_t)m * H + n] = acc[r] + agg[(size_t)m * H + n] * rc + bias[n];
  }
}

// ---------------------------------------------------------------- BN stats
__global__ void k_bnstats(const float* __restrict__ x_new, float* __restrict__ mu,
                          float* __restrict__ rstd) {
  __shared__ float s_sum[256], s_sq[256];
  int c = blockIdx.x, t = threadIdx.x;
  float s = 0.f, q = 0.f;
  for (int r = t; r < NNODES; r += 256) {
    float v = x_new[(size_t)r * H + c];
    s += v; q += v * v;
  }
  s_sum[t] = s; s_sq[t] = q;
  __syncthreads();
  for (int off = 128; off > 0; off >>= 1) {
    if (t < off) { s_sum[t] += s_sum[t + off]; s_sq[t] += s_sq[t + off]; }
    __syncthreads();
  }
  if (t == 0) {
    float mval = s_sum[0] / (float)NNODES;
    float vval = s_sq[0] / (float)NNODES - mval * mval;
    mu[c] = mval;
    rstd[c] = rsqrtf(vval + BN_EPS);
  }
}

// ---------------------------------------------------------------- BN apply + relu + residual
__global__ void k_bnapply(float* __restrict__ h, const float* __restrict__ x_new,
                          const float* __restrict__ mu, const float* __restrict__ rstd,
                          const float* __restrict__ gamma, const float* __restrict__ beta) {
  int gid = blockIdx.x * blockDim.x + threadIdx.x;
  int c = gid & 127;
  float v = (x_new[gid] - mu[c]) * rstd[c] * gamma[c] + beta[c];
  h[gid] += fmaxf(v, 0.0f);
}

// ---------------------------------------------------------------- global mean pool
__global__ void k_pool(const int* __restrict__ batch, const float* __restrict__ h,
                       float* __restrict__ gsum, float* __restrict__ gcnt) {
  int gid = blockIdx.x * blockDim.x + threadIdx.x;  // N*H
  int n = gid >> 7, c = gid & 127;
  int g = batch[n];
  atomicAdd(&gsum[(size_t)g * H + c], h[gid]);
  if (c == 0) atomicAdd(&gcnt[g], 1.0f);
}

// ---------------------------------------------------------------- classifier
__global__ void k_classifier(const float* __restrict__ gsum, const float* __restrict__ gcnt,
                             const float* __restrict__ W1, const float* __restrict__ b1,
                             const float* __restrict__ W2, const float* __restrict__ b2,
                             float* __restrict__ out) {
  __shared__ float s_g[H];
  __shared__ float s_p[64];
  int g = blockIdx.x, t = threadIdx.x;  // 64 threads
  float rc = 1.0f / fmaxf(gcnt[g], 1.0f);
  for (int i = t; i < H; i += 64) s_g[i] = gsum[(size_t)g * H + i] * rc;
  __syncthreads();
  float acc = b1[t];
  for (int i = 0; i < H; ++i) acc += s_g[i] * W1[i * 64 + t];
  acc = fmaxf(acc, 0.0f);
  s_p[t] = acc * W2[t];
  __syncthreads();
  for (int off = 32; off > 0; off >>= 1) {
    if (t < off) s_p[t] += s_p[t + off];
    __syncthreads();
  }
  if (t == 0) out[g] = s_p[0] + b2[0];
}

// ---------------------------------------------------------------- launcher
extern "C" void kernel_launch(void* const* d_in, const int* in_sizes, int n_in,
                              void* d_out, int out_size, void* d_ws, size_t ws_size,
                              hipStream_t stream) {
  const float* x   = (const float*)d_in[0];
  const int*  eidx = (const int*)d_in[1];
  const float* ea  = (const float*)d_in[2];
  const int*  batch= (const int*)d_in[3];
  const float* nW  = (const float*)d_in[4];
  const float* nb  = (const float*)d_in[5];
  const float* eW  = (const float*)d_in[6];
  const float* eb  = (const float*)d_in[7];
  const float* rW  = (const float*)d_in[8];
  const float* cb  = (const float*)d_in[9];
  const float* gam = (const float*)d_in[10];
  const float* bet = (const float*)d_in[11];
  const float* W1  = (const float*)d_in[12];
  const float* b1  = (const float*)d_in[13];
  const float* W2  = (const float*)d_in[14];
  const float* b2  = (const float*)d_in[15];
  float* out = (float*)d_out;

  float* ws    = (float*)d_ws;
  float* h     = ws;                   // 2048*128
  float* msg   = h + NNODES * H;       // 8192*128
  float* agg   = msg + NEDGES * H;     // 2048*128
  float* cnt   = agg + NNODES * H;     // 2048  (contiguous after agg for one memset)
  float* x_new = cnt + NNODES;         // 2048*128
  float* mu    = x_new + NNODES * H;   // 128
  float* rstd  = mu + H;               // 128
  float* gsum  = rstd + H;             // 128*128
  float* gcnt  = gsum + NGRAPH * H;    // 128 (contiguous after gsum)

  k_node_embed<<<NNODES / 16, 256, 0, stream>>>(x, nW, nb, h);

  for (int l = 0; l < LAYERS; ++l) {
    (void)hipMemsetAsync(agg, 0, (size_t)(NNODES * H + NNODES) * sizeof(float), stream);
    k_msg<<<NEDGES / 64, 256, 0, stream>>>(h, eidx, ea,
                                           eW + (size_t)l * EDIM * H * H,
                                           eb + (size_t)l * H * H, msg);
    k_scatter_msg<<<NEDGES * H / 256, 256, 0, stream>>>(eidx + NEDGES, msg, agg, cnt);
    k_update<<<NNODES / 16, 256, 0, stream>>>(h, rW + (size_t)l * H * H, agg, cnt,
                                              cb + l * H, x_new);
    k_bnstats<<<H, 256, 0, stream>>>(x_new, mu, rstd);
    k_bnapply<<<NNODES * H / 256, 256, 0, stream>>>(h, x_new, mu, rstd,
                                                    gam + l * H, bet + l * H);
  }

  (void)hipMemsetAsync(gsum, 0, (size_t)(NGRAPH * H + NGRAPH) * sizeof(float), stream);
  k_pool<<<NNODES * H / 256, 256, 0, stream>>>(batch, h, gsum, gcnt);
  k_classifier<<<NGRAPH, 64, 0, stream>>>(gsum, gcnt, W1, b1, W2, b2, out);
}